// TorchSparseVoxelBackbone8x_13134009991789
// MI455X (gfx1250) — compile-verified
//
#include <hip/hip_runtime.h>

// ---------------------------------------------------------------------------
// Sparse 3D conv backbone (TorchSparse-style) for gfx1250 / MI455X.
//   Layer l: conv (gather + WMMA f32 GEMM) -> BN stats -> BN apply + ReLU.
// fp32 path uses V_WMMA_F32_16X16X4_F32 (native CDNA5 fp32 matrix op).
// ---------------------------------------------------------------------------

typedef float v2f __attribute__((ext_vector_type(2)));
typedef float v8f __attribute__((ext_vector_type(8)));

#define BN_EPS 1e-3f

// ---------------------------------------------------------------------------
// Gather-GEMM sparse conv.
//   Wave owns a 16-row output tile and ALL co columns (CO/16 accumulators),
//   so each gathered A tile is reused across every co-tile.
//
// WMMA f32 16x16x4 per-lane layout (wave32):
//   A (16x4, MxK): lane&15 = M row; lane>=16 selects K={2,3} else K={0,1}
//                  -> float2 {A[M][kOff], A[M][kOff+1]}
//   B (4x16, KxN): lane&15 = N col; same half-wave K selection
//   C/D (16x16):   VGPR r: lanes 0-15 -> M=r, lanes 16-31 -> M=r+8; N=lane&15
// ---------------------------------------------------------------------------
template <int CI, int CO, int KK>
__global__ __launch_bounds__(256) void spconv_wmma_kernel(
    const float* __restrict__ feat,   // [Nin, CI]
    const int*   __restrict__ rb,     // [KK, N]
    const float* __restrict__ W,      // [KK, CI, CO]
    float*       __restrict__ out,    // [N, CO]
    int N)
{
  constexpr int NT = CO / 16;                 // co tiles per wave
  const int lane  = threadIdx.x & 31;
  const int wave  = threadIdx.x >> 5;
  const int mBase = (blockIdx.x * 8 + wave) * 16;
  if (mBase >= N) return;                     // wave-uniform exit: EXEC stays all-1s

  const int r15  = lane & 15;
  const int hi   = lane >> 4;                 // 0: K pair {0,1}; 1: K pair {2,3}
  const int kOff = hi * 2;
  const int row  = mBase + r15;

  v8f acc[NT];
#pragma unroll
  for (int t = 0; t < NT; ++t) acc[t] = {};

  for (int k = 0; k < KK; ++k) {
    // rulebook gather index for this lane's row (rows >= N behave like idx=-1)
    int idx = (row < N) ? rb[(long)k * N + row] : -1;
    const float msk = (idx >= 0) ? 1.0f : 0.0f;
    const float* __restrict__ arow = feat + (long)(idx < 0 ? 0 : idx) * CI;
    const float* __restrict__ wk   = W + (long)k * CI * CO;

#pragma unroll
    for (int cb = 0; cb < CI; cb += 4) {
      // A tile: aligned 8-byte gather, masked for invalid neighbors
      v2f a = *(const v2f*)(arow + cb + kOff);
      a *= msk;
#pragma unroll
      for (int t = 0; t < NT; ++t) {
        v2f b;
        b.x = wk[(cb + kOff    ) * CO + t * 16 + r15];
        b.y = wk[(cb + kOff + 1) * CO + t * 16 + r15];
        acc[t] = __builtin_amdgcn_wmma_f32_16x16x4_f32(
            false, a, false, b, (short)0, acc[t], false, false);
      }
    }
  }

  // scatter C/D tiles to [N, CO]
#pragma unroll
  for (int t = 0; t < NT; ++t) {
#pragma unroll
    for (int r = 0; r < 8; ++r) {
      int orow = mBase + r + hi * 8;
      if (orow < N) out[(long)orow * CO + t * 16 + r15] = acc[t][r];
    }
  }
}

// ---------------------------------------------------------------------------
// BN stats: per-channel sum and sum-of-squares.
//   Block-local LDS accumulation (ds_add_f32) then one global atomic per
//   channel per block. stats layout: [sum(co) | sumsq(co)]. co is a power of 2.
// ---------------------------------------------------------------------------
__global__ __launch_bounds__(256) void bn_stats_kernel(
    const float* __restrict__ x, float* __restrict__ stats,
    int N, int co, int rowsPerBlock)
{
  __shared__ float s_sum[128];
  __shared__ float s_ss[128];
  const int tid = threadIdx.x;
  if (tid < 128) { s_sum[tid] = 0.0f; s_ss[tid] = 0.0f; }
  __syncthreads();

  long total = (long)N * co;
  long start = (long)blockIdx.x * rowsPerBlock * co;
  long end   = start + (long)rowsPerBlock * co;
  if (end > total) end = total;

  for (long i = start + tid; i < end; i += 256) {
    float v = x[i];
    int c = (int)(i & (long)(co - 1));
    atomicAdd(&s_sum[c], v);
    atomicAdd(&s_ss[c], v * v);
  }
  __syncthreads();

  if (tid < co) {
    atomicAdd(&stats[tid],      s_sum[tid]);
    atomicAdd(&stats[co + tid], s_ss[tid]);
  }
}

// ---------------------------------------------------------------------------
// BN apply + ReLU (training-mode BN, biased var, eps = 1e-3).
// ---------------------------------------------------------------------------
__global__ __launch_bounds__(256) void bn_apply_kernel(
    const float* __restrict__ x, const float* __restrict__ stats,
    const float* __restrict__ gamma, const float* __restrict__ beta,
    float* __restrict__ y, int N, int co)
{
  long i = (long)blockIdx.x * blockDim.x + threadIdx.x;
  long total = (long)N * co;
  if (i >= total) return;
  int c = (int)(i & (long)(co - 1));
  float invN = 1.0f / (float)N;
  float mu   = stats[c] * invN;
  float var  = stats[co + c] * invN - mu * mu;
  float sc   = gamma[c] * rsqrtf(var + BN_EPS);
  float v    = (x[i] - mu) * sc + beta[c];
  y[i] = v > 0.0f ? v : 0.0f;
}

// ---------------------------------------------------------------------------
// Host-side dispatch
// ---------------------------------------------------------------------------
static void launch_conv(int layer, const float* feat, const int* rb,
                        const float* W, float* out, int N, hipStream_t s)
{
  dim3 blk(256);
  dim3 grd((unsigned)((N + 127) / 128));
  switch (layer) {
    case 0:  spconv_wmma_kernel<4,   16, 27><<<grd, blk, 0, s>>>(feat, rb, W, out, N); break;
    case 1:  spconv_wmma_kernel<16,  16, 27><<<grd, blk, 0, s>>>(feat, rb, W, out, N); break;
    case 2:  spconv_wmma_kernel<16,  32, 27><<<grd, blk, 0, s>>>(feat, rb, W, out, N); break;
    case 3:
    case 4:  spconv_wmma_kernel<32,  32, 27><<<grd, blk, 0, s>>>(feat, rb, W, out, N); break;
    case 5:  spconv_wmma_kernel<32,  64, 27><<<grd, blk, 0, s>>>(feat, rb, W, out, N); break;
    case 6:
    case 7:  spconv_wmma_kernel<64,  64, 27><<<grd, blk, 0, s>>>(feat, rb, W, out, N); break;
    case 8:  spconv_wmma_kernel<64, 128, 27><<<grd, blk, 0, s>>>(feat, rb, W, out, N); break;
    case 9:
    case 10: spconv_wmma_kernel<128,128, 27><<<grd, blk, 0, s>>>(feat, rb, W, out, N); break;
    case 11: spconv_wmma_kernel<128,128,  3><<<grd, blk, 0, s>>>(feat, rb, W, out, N); break;
    default: break;
  }
}

extern "C" void kernel_launch(void* const* d_in, const int* in_sizes, int n_in,
                              void* d_out, int out_size, void* d_ws, size_t ws_size,
                              hipStream_t stream)
{
  (void)n_in; (void)out_size; (void)ws_size;

  // Input layout (setup_inputs dict order, lists flattened):
  //   0: voxel_features [N0, 4] f32
  //   1: coords (unused)   2: batch_size (unused)
  //   3..14:  weights[l]  [K, ci, co] f32
  //   15..26: gammas[l]   [co] f32
  //   27..38: betas[l]    [co] f32
  //   39..50: rulebooks[l][K, N_out] i32
  static const int KKs[12] = {27,27,27,27,27,27,27,27,27,27,27,3};
  static const int COs[12] = {16,16,32,32,32,64,64,64,128,128,128,128};

  const float* feats = (const float*)d_in[0];
  const float* W[12]; const float* G[12]; const float* Bt[12]; const int* RB[12];
  int Nout[12];
  for (int l = 0; l < 12; ++l) {
    W[l]  = (const float*)d_in[3  + l];
    G[l]  = (const float*)d_in[15 + l];
    Bt[l] = (const float*)d_in[27 + l];
    RB[l] = (const int*)  d_in[39 + l];
    Nout[l] = in_sizes[39 + l] / KKs[l];
  }

  // Workspace: two ping-pong feature buffers + a small BN-stats buffer.
  long bufElems = 0;
  for (int l = 0; l < 12; ++l) {
    long e = (long)Nout[l] * COs[l];
    if (e > bufElems) bufElems = e;
  }
  float* bufA  = (float*)d_ws;
  float* bufB  = bufA + bufElems;
  float* stats = bufB + bufElems;     // up to 2*128 floats

  const float* cur = feats;
  for (int l = 0; l < 12; ++l) {
    float* convOut = (l & 1) ? bufB : bufA;
    const int N  = Nout[l];
    const int co = COs[l];

    launch_conv(l, cur, RB[l], W[l], convOut, N, stream);

    hipMemsetAsync(stats, 0, 2 * co * sizeof(float), stream);
    const int rowsPerBlock = 512;
    unsigned sblocks = (unsigned)((N + rowsPerBlock - 1) / rowsPerBlock);
    bn_stats_kernel<<<dim3(sblocks), 256, 0, stream>>>(convOut, stats, N, co, rowsPerBlock);

    long total = (long)N * co;
    float* dst = (l == 11) ? (float*)d_out : convOut;   // final layer -> d_out
    bn_apply_kernel<<<dim3((unsigned)((total + 255) / 256)), 256, 0, stream>>>(
        convOut, stats, G[l], Bt[l], dst, N, co);

    cur = convOut;  // BN applied in-place; next layer reads normalized features
  }
}